// SetLoss_30760555773959
// MI455X (gfx1250) — compile-verified
//
#include <hip/hip_runtime.h>
#include <hip/hip_bf16.h>

typedef __attribute__((ext_vector_type(2))) float v2f;
typedef __attribute__((ext_vector_type(8))) float v8f;

#define NBLOCKS 512   // must be multiple of 4 for the WMMA finisher
#define NTHREADS 256  // 8 waves (wave32)

// ---------------------------------------------------------------------------
// Stage 1: grid-stride streaming reduction over N rows.
// Accumulates per-thread: [0]=sum(-score) [1]=sum(d.d) [2]=sum succ&label
// [3]=cnt label  [4]=sum succ&pred  [5]=cnt pred  [6]=sum per&pred  [7]=0
// Wave32 shuffle tree -> LDS -> one 16-float padded partial per block in ws.
// ---------------------------------------------------------------------------
__global__ void setloss_partials(const float* __restrict__ scores,
                                 const float* __restrict__ cov,
                                 const float* __restrict__ pos,
                                 const float* __restrict__ pose,
                                 const int*   __restrict__ indices,
                                 const int*   __restrict__ match,
                                 float* __restrict__ partials,
                                 int N, int n, int m) {
    float acc[8];
#pragma unroll
    for (int k = 0; k < 8; ++k) acc[k] = 0.0f;

    const int np1 = n + 1;
    const int mp1 = m + 1;
    const int stride = gridDim.x * blockDim.x;

    for (int i = blockIdx.x * blockDim.x + threadIdx.x; i < N; i += stride) {
        // prefetch next grid-stride pose row (widest-stride stream) ->
        // global_prefetch_b8 on gfx1250
        if (i + stride < N) __builtin_prefetch(&pose[7 * (size_t)(i + stride)], 0, 1);

        const int b = i / n;            // group
        const int r = i - b * n;        // row within group (drops dustbin row)

        const int mi = match[i];
        const int oi = indices[i];
        const bool label_valid = (mi != -1);
        const bool pred_valid  = (oi != -1);
        const int  li = label_valid ? mi : m;     // route -1 to dustbin col m

        // gather one score: scores[b, r, li]
        const size_t sidx = ((size_t)b * np1 + r) * mp1 + li;
        acc[0] += -scores[sidx];

        const float succ = (oi == mi) ? 1.0f : 0.0f;
        acc[2] += label_valid ? succ : 0.0f;
        acc[3] += label_valid ? 1.0f : 0.0f;
        acc[4] += pred_valid ? succ : 0.0f;
        acc[5] += pred_valid ? 1.0f : 0.0f;

        const size_t i3 = 3 * (size_t)i;
        const size_t i7 = 7 * (size_t)i;
        const float dx = pos[i3 + 0] - pose[i7 + 0];
        const float dy = pos[i3 + 1] - pose[i7 + 1];
        const float dz = pos[i3 + 2] - pose[i7 + 2];
        const float dd = dx * dx + dy * dy + dz * dz;
        acc[1] += dd;

        const float c = cov[i];
        const float per = dd / c + 0.12f * __logf(c);  // dd/c + 3*0.04*log(c)
        acc[6] += pred_valid ? per : 0.0f;
    }

    // wave32 tree reduction
#pragma unroll
    for (int off = 16; off > 0; off >>= 1) {
#pragma unroll
        for (int k = 0; k < 8; ++k) acc[k] += __shfl_down(acc[k], off, 32);
    }

    __shared__ float red[NTHREADS / 32][8];
    const int wave = threadIdx.x >> 5;
    const int lane = threadIdx.x & 31;
    if (lane == 0) {
#pragma unroll
        for (int k = 0; k < 8; ++k) red[wave][k] = acc[k];
    }
    __syncthreads();

    if (threadIdx.x == 0) {
        float s[8];
#pragma unroll
        for (int k = 0; k < 8; ++k) s[k] = 0.0f;
        for (int w = 0; w < NTHREADS / 32; ++w)
#pragma unroll
            for (int k = 0; k < 8; ++k) s[k] += red[w][k];
        float* p = partials + (size_t)blockIdx.x * 16;
#pragma unroll
        for (int k = 0; k < 8; ++k) p[k] = s[k];
#pragma unroll
        for (int k = 8; k < 16; ++k) p[k] = 0.0f;   // pad M-rows 8..15 with 0
    }
}

// ---------------------------------------------------------------------------
// Stage 2: single wave32. Reduce NBLOCKS partial vectors with
// V_WMMA_F32_16X16X4_F32: A[m][k] = partial[g+k][m] (m = component, rows 8-15
// zero-padded), B = ones(4x16)  =>  D[m][j] += sum_k partial[g+k][m].
// ISA A-layout (32-bit 16x4): VGPR0 lanes0-15 -> K=0, lanes16-31 -> K=2;
//                             VGPR1 lanes0-15 -> K=1, lanes16-31 -> K=3.
// ISA C/D-layout: lane j holds N=j; VGPR v holds M=v (lanes 0-15).
// So after the loop, lane 0 VGPRs 0..7 hold the 8 component sums.
// EXEC is all ones: 32-thread launch, no divergence before the WMMA loop.
// ---------------------------------------------------------------------------
__global__ void setloss_finalize(const float* __restrict__ partials,
                                 float* __restrict__ out,
                                 int G, float invN) {
    const int lane = threadIdx.x;          // 0..31
    const int half = lane >> 4;            // 0: K={0,1}   1: K={2,3}
    const int mrow = lane & 15;            // M row (component index, 8-15 pad)

    v8f c = {};
    v2f bones;
    bones[0] = 1.0f;
    bones[1] = 1.0f;

    for (int g = 0; g < G; g += 4) {
        v2f a;
        const int v0 = g + half * 2;       // partial index for this lane's K
        a[0] = partials[(size_t)v0 * 16 + mrow];
        a[1] = partials[(size_t)(v0 + 1) * 16 + mrow];
        c = __builtin_amdgcn_wmma_f32_16x16x4_f32(
                /*neg_a=*/false, a, /*neg_b=*/false, bones,
                /*c_mod=*/(short)0, c, /*reuse_a=*/false, /*reuse_b=*/false);
    }

    if (lane == 0) {
        const float sum_cost = c[0];
        const float sum_dd   = c[1];
        const float s_sl     = c[2];
        const float cnt_l    = c[3];
        const float s_sp     = c[4];
        const float cnt_p    = c[5];
        const float s_per    = c[6];

        const float loss_match = sum_cost * invN;        // mean over N
        const float loss_pos   = sum_dd * invN;          // mean(d*d)*3 = sum/N
        const float recall     = (cnt_l > 0.0f) ? s_sl / cnt_l : 1.0f;
        const float precision  = (cnt_p > 0.0f) ? s_sp / cnt_p : 1.0f;
        const float loss_cov   = (cnt_p > 0.0f) ? s_per / cnt_p : 5.0f;
        const float total = loss_match + loss_pos + 0.1f * loss_cov;

        out[0] = total;
        out[1] = loss_match;
        out[2] = loss_pos;
        out[3] = 0.0f;        // loss_rot
        out[4] = loss_cov;
        out[5] = precision;
        out[6] = recall;
    }
}

extern "C" void kernel_launch(void* const* d_in, const int* in_sizes, int n_in,
                              void* d_out, int out_size, void* d_ws, size_t ws_size,
                              hipStream_t stream) {
    // setup_inputs dict order: scores, cov, pos, pose, indices, match
    const float* scores  = (const float*)d_in[0];
    const float* cov     = (const float*)d_in[1];
    const float* pos     = (const float*)d_in[2];
    const float* pose    = (const float*)d_in[3];
    const int*   indices = (const int*)d_in[4];
    const int*   match   = (const int*)d_in[5];
    float* out = (float*)d_out;

    const int N = in_sizes[5];                       // flattened rows
    const int n = 31;                                // rows per group (ref)
    const int B = N / n;                             // groups
    const int mp1 = (int)((long long)in_sizes[0] / ((long long)B * (n + 1)));
    const int m = mp1 - 1;                           // derived from scores size

    float* partials = (float*)d_ws;                  // NBLOCKS * 16 floats

    setloss_partials<<<NBLOCKS, NTHREADS, 0, stream>>>(
        scores, cov, pos, pose, indices, match, partials, N, n, m);

    setloss_finalize<<<1, 32, 0, stream>>>(partials, out, NBLOCKS,
                                           1.0f / (float)N);
}